// MGCN_37744172597277
// MI455X (gfx1250) — compile-verified
//
#include <hip/hip_runtime.h>
#include <hip/hip_bf16.h>
#include <math.h>

#define NE    50000
#define DM    200
#define NEDGE 200000
#define BATCH 1024
#define NBAS  4
#define NF    96
#define FLATD 18816
#define EPSB  1e-5f
#define KC    64          // K-chunk depth for TDM-staged GEMM (multiple of 4)
#define BROW  48          // padded LDS row stride (32 data + 16 pad dwords)

typedef float v2f __attribute__((ext_vector_type(2)));
typedef float v8f __attribute__((ext_vector_type(8)));
typedef unsigned int u32x4 __attribute__((ext_vector_type(4)));
typedef int i32x4 __attribute__((ext_vector_type(4)));
typedef int i32x8 __attribute__((ext_vector_type(8)));

// ---------------------------------------------------------------------------
// Tensor Data Mover: async 2D tile load Global -> LDS. Builds the D# per
// CDNA5 ISA ch.8: group0 = {count/lds_addr/global_addr/type}, group1 =
// {data_size, pad cfg, tensor_dim0/1, tile_dim0/1, stride0}. OOB elements
// (beyond tensor_dim0/1) are zero-filled by the TDM. Tracked by TENSORcnt.
// This toolchain's builtin takes 6 args (extra zero group before cpol).
// ---------------------------------------------------------------------------
__device__ inline void tdm_load_2d(const void* gaddr, unsigned lds_byte,
                                   unsigned dim0, unsigned dim1,
                                   unsigned tile0, unsigned tile1,
                                   unsigned stride0,
                                   unsigned pad_en, unsigned pad_int,
                                   unsigned pad_amt)
{
    unsigned long long ga = (unsigned long long)(size_t)gaddr;
    u32x4 g0;
    g0.x = 1u;                                          // count=1, user desc
    g0.y = lds_byte;                                    // LDS byte address
    g0.z = (unsigned)ga;                                // global_addr[31:0]
    g0.w = (unsigned)((ga >> 32) & 0x01FFFFFFull) | (2u << 30); // addr[56:32]|type=2
    i32x8 g1;
    g1[0] = (int)((2u << 16) | (pad_en << 20) | (pad_int << 22) | (pad_amt << 25));
    g1[1] = (int)((dim0 & 0xFFFFu) << 16);              // tensor_dim0[15:0]
    g1[2] = (int)((dim0 >> 16) | ((dim1 & 0xFFFFu) << 16));
    g1[3] = (int)((dim1 >> 16) | (tile0 << 16));        // tile_dim0
    g1[4] = (int)(tile1 & 0xFFFFu);                     // tile_dim1 (tile2=0)
    g1[5] = (int)stride0;                               // tensor_dim0_stride lo
    g1[6] = 0;
    g1[7] = 0;
    i32x4 z4 = {0, 0, 0, 0};
    i32x8 z8 = {0, 0, 0, 0, 0, 0, 0, 0};
    __builtin_amdgcn_tensor_load_to_lds(g0, g1, z4, z4, z8, 0);
}

__device__ inline unsigned lds_offset_of(const void* p)
{
    // generic LDS address = {shared aperture, offset}; low 32 bits = offset
    return (unsigned)(unsigned long long)p;
}

// ---------------------------------------------------------------------------
// NN GEMM: C[M,Nn] (+=|=) A[M,lda-slab] * B[K,Nn]. Block = 4 waves sharing one
// 32-wide B panel (wave w -> M-tile mg*4+w). B panels are TDM-loaded into a
// double-buffered, pad-destrided LDS region (bank-conflict-free ds reads);
// each wave runs V_WMMA_F32_16X16X4_F32 on a 16x32 C tile. K-split slabs
// (blockIdx.y) accumulate with atomics.
// ---------------------------------------------------------------------------
__global__ __launch_bounds__(128) void gemm_nn_kernel(
    const float* __restrict__ A, const float* __restrict__ B,
    const float* __restrict__ bias, float* __restrict__ C,
    int M, int lda, int Nn, int kcount, int accumulate)
{
    __shared__ float bpanel[2][KC * BROW];
    int mtiles = M >> 4;
    int npairs = (Nn + 31) / 32;
    int mg = blockIdx.x / npairs;
    int np = blockIdx.x % npairs;
    int slabk = (int)blockIdx.y * kcount;

    int wid  = threadIdx.x >> 5;
    int mt   = mg * 4 + wid;
    bool active = mt < mtiles;
    int lane = threadIdx.x & 31;
    int lm   = lane & 15;
    int lk   = (lane >> 4) << 1;                 // K sub-offset 0 or 2
    int n0p  = np * 32;
    int n0   = n0p + lm, n1 = n0 + 16;
    bool v0n = n0 < Nn, v1n = n1 < Nn;

    const float* Arow = A + (size_t)((active ? mt * 16 : 0) + lm) * lda + slabk;
    unsigned lds0 = lds_offset_of(&bpanel[0][0]);
    unsigned lds1 = lds_offset_of(&bpanel[1][0]);

    v8f acc0 = {0.f,0.f,0.f,0.f,0.f,0.f,0.f,0.f};
    v8f acc1 = {0.f,0.f,0.f,0.f,0.f,0.f,0.f,0.f};

    int nchunks = (kcount + KC - 1) / KC;
    if (wid == 0) {
        int r0 = kcount < KC ? kcount : KC;
        tdm_load_2d(B + (size_t)slabk * Nn + n0p, lds0,
                    (unsigned)(Nn - n0p), (unsigned)r0, 32u, (unsigned)r0,
                    (unsigned)Nn, 1u, 4u, 15u);   // pad 16 dw every 32 -> stride 48
    }
    for (int c = 0; c < nchunks; ++c) {
        if (wid == 0) __builtin_amdgcn_s_wait_tensorcnt(0);
        __syncthreads();                          // panel c visible to all waves
        if (wid == 0 && c + 1 < nchunks) {        // prefetch next panel (other buf)
            int k0n = (c + 1) * KC;
            int rn  = kcount - k0n; if (rn > KC) rn = KC;
            tdm_load_2d(B + (size_t)(slabk + k0n) * Nn + n0p,
                        ((c + 1) & 1) ? lds1 : lds0,
                        (unsigned)(Nn - n0p), (unsigned)rn, 32u, (unsigned)rn,
                        (unsigned)Nn, 1u, 4u, 15u);
        }
        const float* bp = bpanel[c & 1];
        const float* Ac = Arow + c * KC;
        int rc = kcount - c * KC; if (rc > KC) rc = KC;
        for (int k = 0; k < rc; k += 4) {
            int kk = k + lk;
            v2f a = *(const v2f*)(Ac + kk);
            v2f b0, b1;
            b0.x = bp[kk * BROW + lm];
            b0.y = bp[(kk + 1) * BROW + lm];
            b1.x = bp[kk * BROW + lm + 16];
            b1.y = bp[(kk + 1) * BROW + lm + 16];
            acc0 = __builtin_amdgcn_wmma_f32_16x16x4_f32(false, a, false, b0,
                                                         (short)0, acc0, false, false);
            acc1 = __builtin_amdgcn_wmma_f32_16x16x4_f32(false, a, false, b1,
                                                         (short)0, acc1, false, false);
        }
    }

    if (!active) return;                          // wave-uniform
    int row0 = mt * 16 + ((lane >> 4) << 3);
    if (accumulate) {
#pragma unroll
        for (int j = 0; j < 8; ++j) {
            size_t r = (size_t)(row0 + j) * Nn;
            if (v0n) unsafeAtomicAdd(&C[r + n0], acc0[j]);
            if (v1n) unsafeAtomicAdd(&C[r + n1], acc1[j]);
        }
    } else {
        float bi0 = (bias && v0n) ? bias[n0] : 0.f;
        float bi1 = (bias && v1n) ? bias[n1] : 0.f;
#pragma unroll
        for (int j = 0; j < 8; ++j) {
            size_t r = (size_t)(row0 + j) * Nn;
            if (v0n) C[r + n0] = acc0[j] + bi0;
            if (v1n) C[r + n1] = acc1[j] + bi1;
        }
    }
}

// ---------------------------------------------------------------------------
// Logits: Out[B,NE] = sigmoid(xi2[B,200] @ h[NE,200]^T + ent_bias). One TDM
// loads a 32-row h panel (zero-filled past NE) into LDS, shared by 4 waves.
// ---------------------------------------------------------------------------
__global__ __launch_bounds__(128) void logits_kernel(
    const float* __restrict__ A, const float* __restrict__ Hm,
    const float* __restrict__ ebias, float* __restrict__ Out)
{
    __shared__ float hpanel[32 * DM];
    const int npairs = (NE + 31) / 32;            // 1563
    int mg = blockIdx.x / npairs;
    int np = blockIdx.x % npairs;
    int wid  = threadIdx.x >> 5;
    int mt   = mg * 4 + wid;                      // < 64 always
    int lane = threadIdx.x & 31;
    int lm   = lane & 15;
    int lk   = (lane >> 4) << 1;
    int n0p  = np * 32;

    if (wid == 0) {
        tdm_load_2d(Hm + (size_t)n0p * DM, lds_offset_of(&hpanel[0]),
                    (unsigned)DM, (unsigned)(NE - n0p), (unsigned)DM, 32u,
                    (unsigned)DM, 0u, 0u, 0u);
        __builtin_amdgcn_s_wait_tensorcnt(0);
    }
    __syncthreads();

    const float* Arow = A + (size_t)(mt * 16 + lm) * DM;
    v8f acc0 = {0.f,0.f,0.f,0.f,0.f,0.f,0.f,0.f};
    v8f acc1 = {0.f,0.f,0.f,0.f,0.f,0.f,0.f,0.f};
    for (int k = 0; k < DM; k += 4) {
        int kk = k + lk;
        v2f a  = *(const v2f*)(Arow + kk);
        v2f b0 = *(const v2f*)&hpanel[lm * DM + kk];
        v2f b1 = *(const v2f*)&hpanel[(lm + 16) * DM + kk];
        acc0 = __builtin_amdgcn_wmma_f32_16x16x4_f32(false, a, false, b0,
                                                     (short)0, acc0, false, false);
        acc1 = __builtin_amdgcn_wmma_f32_16x16x4_f32(false, a, false, b1,
                                                     (short)0, acc1, false, false);
    }

    int n0 = n0p + lm, n1 = n0 + 16;
    bool v0n = n0 < NE, v1n = n1 < NE;
    int row0 = mt * 16 + ((lane >> 4) << 3);
    float eb0 = v0n ? ebias[n0] : 0.f;
    float eb1 = v1n ? ebias[n1] : 0.f;
#pragma unroll
    for (int j = 0; j < 8; ++j) {
        size_t r = (size_t)(row0 + j) * NE;
        if (v0n) Out[r + n0] = 1.f / (1.f + __expf(-(acc0[j] + eb0)));
        if (v1n) Out[r + n1] = 1.f / (1.f + __expf(-(acc1[j] + eb1)));
    }
}

// ---------------------------------------------------------------------------
// Per-edge message + scatter: h[dst] += norm * sum_b att[rt,b] * Y[b][src].
// ---------------------------------------------------------------------------
__global__ __launch_bounds__(200) void msg_kernel(
    const float* __restrict__ Y, float* __restrict__ Hm,
    const int* __restrict__ eidx, const int* __restrict__ etype,
    const float* __restrict__ enorm, const float* __restrict__ att)
{
    int e  = blockIdx.x;
    int s  = eidx[e];
    int dn = eidx[NEDGE + e];
    int rt = etype[e];
    float nr = enorm[e];
    float c0 = att[rt * NBAS + 0] * nr;
    float c1 = att[rt * NBAS + 1] * nr;
    float c2 = att[rt * NBAS + 2] * nr;
    float c3 = att[rt * NBAS + 3] * nr;
    int d = threadIdx.x;
    const float* ys = Y + (size_t)s * DM + d;
    const size_t stride = (size_t)NE * DM;
    float v = c0 * ys[0] + c1 * ys[stride] + c2 * ys[2 * stride] + c3 * ys[3 * stride];
    unsafeAtomicAdd(&Hm[(size_t)dn * DM + d], v);
}

// ---------------------------------------------------------------------------
// Build stacked [B,20,20] image (pre-bn0) + bn0 batch statistics.
// ---------------------------------------------------------------------------
__global__ __launch_bounds__(400) void stack_kernel(
    const float* __restrict__ Hm, const float* __restrict__ rel_table,
    const int* __restrict__ srcB, const int* __restrict__ relB,
    float* __restrict__ xi, float* __restrict__ stats)
{
    __shared__ float ss[512], sq[512];
    int b = blockIdx.x, f = threadIdx.x;
    int d = f >> 1;
    float v = (f & 1) ? rel_table[(size_t)relB[b] * DM + d]
                      : Hm[(size_t)srcB[b] * DM + d];
    xi[b * 400 + f] = v;
    ss[f] = v; sq[f] = v * v;
    if (f < 112) { ss[f + 400] = 0.f; sq[f + 400] = 0.f; }
    __syncthreads();
    for (int s2 = 256; s2 > 0; s2 >>= 1) {
        if (f < s2) { ss[f] += ss[f + s2]; sq[f] += sq[f + s2]; }
        __syncthreads();
    }
    if (f == 0) { unsafeAtomicAdd(&stats[0], ss[0]); unsafeAtomicAdd(&stats[1], sq[0]); }
}

__global__ void bn0_final_kernel(const float* __restrict__ g,
                                 const float* __restrict__ bb,
                                 float* __restrict__ stats)
{
    float cnt = 400.f * (float)BATCH;
    float m   = stats[0] / cnt;
    float var = stats[1] / cnt - m * m;
    float sc  = g[0] * rsqrtf(var + EPSB);
    stats[2] = sc;
    stats[3] = bb[0] - m * sc;
}

// ---------------------------------------------------------------------------
// Direct 7x7 VALID conv on bn0-normalized 20x20 image -> [B,96,14,14]+bias,
// with per-channel bn1 batch statistics. Block = (b,f), 196 threads.
// ---------------------------------------------------------------------------
__global__ __launch_bounds__(196) void conv_kernel(
    const float* __restrict__ xi, const float* __restrict__ ck,
    const float* __restrict__ cb, float* __restrict__ out,
    float* __restrict__ stats)
{
    __shared__ float img[400];
    __shared__ float w[49];
    __shared__ float rs[256], rq[256];
    int b = blockIdx.x / NF, f = blockIdx.x % NF;
    int t = threadIdx.x;
    float sc = stats[2], sh = stats[3];
    for (int i = t; i < 400; i += 196) img[i] = xi[b * 400 + i] * sc + sh;
    if (t < 49) w[t] = ck[f * 49 + t];
    __syncthreads();

    int y = t / 14, x = t % 14;
    float acc = cb[f];
#pragma unroll
    for (int i = 0; i < 7; ++i)
#pragma unroll
        for (int j = 0; j < 7; ++j)
            acc += w[i * 7 + j] * img[(y + i) * 20 + (x + j)];
    out[(size_t)b * FLATD + f * 196 + t] = acc;

    rs[t] = acc; rq[t] = acc * acc;
    if (t < 60) { rs[196 + t] = 0.f; rq[196 + t] = 0.f; }
    __syncthreads();
    for (int s2 = 128; s2 > 0; s2 >>= 1) {
        if (t < s2) { rs[t] += rs[t + s2]; rq[t] += rq[t + s2]; }
        __syncthreads();
    }
    if (t == 0) {
        unsafeAtomicAdd(&stats[16 + f], rs[0]);
        unsafeAtomicAdd(&stats[128 + f], rq[0]);
    }
}

__global__ void bn1_final_kernel(const float* __restrict__ g,
                                 const float* __restrict__ bb,
                                 float* __restrict__ stats)
{
    int f = threadIdx.x;
    if (f >= NF) return;
    float cnt = (float)(BATCH * 196);
    float m   = stats[16 + f] / cnt;
    float var = stats[128 + f] / cnt - m * m;
    float sc  = g[f] * rsqrtf(var + EPSB);
    stats[256 + f] = sc;
    stats[384 + f] = bb[f] - m * sc;
}

__global__ __launch_bounds__(196) void bn1_apply_kernel(
    float* __restrict__ out, const float* __restrict__ stats)
{
    int b = blockIdx.x / NF, f = blockIdx.x % NF;
    size_t i = (size_t)b * FLATD + f * 196 + threadIdx.x;
    float v = out[i] * stats[256 + f] + stats[384 + f];
    out[i] = v > 0.f ? v : 0.f;
}

// ---------------------------------------------------------------------------
// bn2 (per-column over 1024 rows) + ReLU; fc bias folded in here.
// ---------------------------------------------------------------------------
__global__ __launch_bounds__(256) void bn2_kernel(
    const float* __restrict__ fc, const float* __restrict__ fcb,
    const float* __restrict__ g, const float* __restrict__ bb,
    float* __restrict__ xi2)
{
    __shared__ float rs[256], rq[256];
    __shared__ float msc[2];
    int c = blockIdx.x, t = threadIdx.x;
    float bi = fcb[c];
    float s = 0.f, q = 0.f;
    for (int r = t; r < BATCH; r += 256) {
        float v = fc[r * DM + c] + bi;
        s += v; q += v * v;
    }
    rs[t] = s; rq[t] = q;
    __syncthreads();
    for (int s2 = 128; s2 > 0; s2 >>= 1) {
        if (t < s2) { rs[t] += rs[t + s2]; rq[t] += rq[t + s2]; }
        __syncthreads();
    }
    if (t == 0) {
        float m   = rs[0] / (float)BATCH;
        float var = rq[0] / (float)BATCH - m * m;
        float sc  = g[c] * rsqrtf(var + EPSB);
        msc[0] = sc; msc[1] = bb[c] - m * sc;
    }
    __syncthreads();
    for (int r = t; r < BATCH; r += 256) {
        float v = (fc[r * DM + c] + bi) * msc[0] + msc[1];
        xi2[r * DM + c] = v > 0.f ? v : 0.f;
    }
}

// ---------------------------------------------------------------------------
extern "C" void kernel_launch(void* const* d_in, const int* in_sizes, int n_in,
                              void* d_out, int out_size, void* d_ws, size_t ws_size,
                              hipStream_t stream)
{
    const float* ent_emb   = (const float*)d_in[0];
    const float* rel_table = (const float*)d_in[1];
    const float* basis     = (const float*)d_in[2];
    const float* att       = (const float*)d_in[3];
    const float* root      = (const float*)d_in[4];
    const float* gbias     = (const float*)d_in[5];
    const float* bn0_g     = (const float*)d_in[6];
    const float* bn0_b     = (const float*)d_in[7];
    const float* conv_k    = (const float*)d_in[8];
    const float* conv_b    = (const float*)d_in[9];
    const float* bn1_g     = (const float*)d_in[10];
    const float* bn1_b     = (const float*)d_in[11];
    const float* fc_W      = (const float*)d_in[12];
    const float* fc_b      = (const float*)d_in[13];
    const float* bn2_g     = (const float*)d_in[14];
    const float* bn2_b     = (const float*)d_in[15];
    const float* ent_bias  = (const float*)d_in[16];
    const float* edge_norm = (const float*)d_in[17];
    // d_in[18] = entity (arange -> identity gather, unused)
    const int*   edge_idx  = (const int*)d_in[19];
    const int*   edge_type = (const int*)d_in[20];
    const int*   srcB      = (const int*)d_in[21];
    const int*   relB      = (const int*)d_in[22];
    float* out = (float*)d_out;
    float* ws  = (float*)d_ws;

    // Workspace layout (floats). Peak = 50e6 floats = 200 MB.
    float* Hm       = ws;                              // [NE*DM]
    float* Y        = ws + (size_t)NE * DM;            // [NBAS*NE*DM]
    // Region below reuses Y's space AFTER msg_kernel completes (stream order).
    float* xi_img   = Y;                               // [B*400]
    float* conv_out = Y + 1000000;                     // [B*18816]
    float* fc_out   = Y + 21000000;                    // [B*200]
    float* xi2      = Y + 22000000;                    // [B*200]
    float* stats    = Y + 23000000;                    // [512]

    // 1. Y[b] = ent_emb @ basis[b];  h = ent_emb @ root + gbias  (TDM+WMMA)
    {
        int mtiles = NE / 16, npairs = (DM + 31) / 32; // 3125, 7
        dim3 grid(((mtiles + 3) / 4) * npairs, 1);     // 782*7
        for (int b = 0; b < NBAS; ++b)
            gemm_nn_kernel<<<grid, 128, 0, stream>>>(
                ent_emb, basis + (size_t)b * DM * DM, nullptr,
                Y + (size_t)b * NE * DM, NE, DM, DM, DM, 0);
        gemm_nn_kernel<<<grid, 128, 0, stream>>>(
            ent_emb, root, gbias, Hm, NE, DM, DM, DM, 0);
    }

    // 2. Relational message passing: scatter-add into h.
    msg_kernel<<<NEDGE, 200, 0, stream>>>(Y, Hm, edge_idx, edge_type,
                                          edge_norm, att);

    // 3. Zero stats + fc accumulator (Y region free now).
    (void)hipMemsetAsync(stats, 0, 512 * sizeof(float), stream);
    (void)hipMemsetAsync(fc_out, 0, (size_t)BATCH * DM * sizeof(float), stream);

    // 4. ConvE: stack image + bn0 stats, finalize bn0.
    stack_kernel<<<BATCH, 400, 0, stream>>>(Hm, rel_table, srcB, relB,
                                            xi_img, stats);
    bn0_final_kernel<<<1, 1, 0, stream>>>(bn0_g, bn0_b, stats);

    // 5. Conv 7x7 (+ bn1 stats), finalize + apply bn1+ReLU.
    conv_kernel<<<BATCH * NF, 196, 0, stream>>>(xi_img, conv_k, conv_b,
                                                conv_out, stats);
    bn1_final_kernel<<<1, 128, 0, stream>>>(bn1_g, bn1_b, stats);
    bn1_apply_kernel<<<BATCH * NF, 196, 0, stream>>>(conv_out, stats);

    // 6. FC: [1024,18816]@[18816,200], K-split x6, atomic accumulate.
    {
        int mtiles = BATCH / 16, npairs = (DM + 31) / 32; // 64, 7
        dim3 grid(((mtiles + 3) / 4) * npairs, 6);        // 16*7 x 6 slabs
        gemm_nn_kernel<<<grid, 128, 0, stream>>>(conv_out, fc_W, nullptr,
                                                 fc_out, BATCH, FLATD, DM,
                                                 FLATD / 6, 1);
    }

    // 7. bn2 + ReLU (fc bias folded in).
    bn2_kernel<<<DM, 256, 0, stream>>>(fc_out, fc_b, bn2_g, bn2_b, xi2);

    // 8. logits = sigmoid(xi2 @ h^T + ent_bias)  (TDM panel + WMMA NT GEMM)
    {
        int npairs = (NE + 31) / 32;                   // 1563
        dim3 grid((BATCH / 16 / 4) * npairs, 1);       // 16*1563
        logits_kernel<<<grid, 128, 0, stream>>>(xi2, Hm, ent_bias, out);
    }
}